// Model_84370337562865
// MI455X (gfx1250) — compile-verified
//
#include <hip/hip_runtime.h>
#include <hip/hip_bf16.h>

#define LEAK 0.1f

typedef float v2f __attribute__((ext_vector_type(2)));
typedef float v8f __attribute__((ext_vector_type(8)));

__device__ __forceinline__ float leaky(float v) { return v >= 0.0f ? v : LEAK * v; }

// ---------------------------------------------------------------------------
// h[n, FOUT] = x[n, FIN] @ W[FIN, FOUT]  (row-major), fp32 via WMMA 16x16x4.
// One wave (32 threads) computes one 16x16 tile of h. FIN % 4 == 0.
// A frag (16x4): lanes 0-15 hold M=lane, K=k0..k0+1 ; lanes 16-31 hold K=k0+2..k0+3
// B frag (4x16): lanes 0-15 hold N=lane, K=k0..k0+1 ; lanes 16-31 hold K=k0+2..k0+3
// C/D (16x16):  VGPR r: lanes 0-15 -> M=r, lanes 16-31 -> M=r+8 ; N = lane&15
// Out-of-range rows/cols: clamp the load index (no exec divergence -> EXEC
// stays all-ones through the WMMAs) and zero the fragment with selects.
// Full tiles (the common case; N % 16 == 0 here) take a branch-free store path
// with compile-time immediate offsets.
// ---------------------------------------------------------------------------
template <int FIN, int FOUT>
__global__ __launch_bounds__(32) void gemm_wmma_f32(
    const float* __restrict__ x, const float* __restrict__ W,
    float* __restrict__ h, int n)
{
    const int lane = threadIdx.x & 31;
    const int half = lane >> 4;     // 0: K pair 0-1, 1: K pair 2-3
    const int l15  = lane & 15;
    const int row0 = blockIdx.x * 16;
    const int col0 = blockIdx.y * 16;
    const int row  = row0 + l15;    // A row this lane loads
    const int col  = col0 + l15;    // B col this lane loads

    const bool rowOK = row < n;
    const bool colOK = col < FOUT;
    const int  rowC  = rowOK ? row : (n - 1);
    const int  colC  = colOK ? col : (FOUT - 1);

    const float* __restrict__ xr = x + (size_t)rowC * FIN;  // A row base
    const float* __restrict__ wc = W + colC;                // B col base

    v8f acc = {};
#pragma unroll
    for (int k0 = 0; k0 < FIN; k0 += 4) {
        const int ka = k0 + half * 2;            // even -> 8B-aligned float2
        v2f a = *reinterpret_cast<const v2f*>(xr + ka);
        a.x = rowOK ? a.x : 0.0f;
        a.y = rowOK ? a.y : 0.0f;
        v2f b;
        b.x = wc[(size_t)(ka    ) * FOUT];
        b.y = wc[(size_t)(ka + 1) * FOUT];
        b.x = colOK ? b.x : 0.0f;
        b.y = colOK ? b.y : 0.0f;
        // (neg_a, A, neg_b, B, c_mod, C, reuse_a, reuse_b)
        acc = __builtin_amdgcn_wmma_f32_16x16x4_f32(
            false, a, false, b, (short)0, acc, false, false);
    }

    if (colOK) {
        if (row0 + 15 < n) {
            // full tile: branch-free stores, constant immediate offsets
            float* __restrict__ hp = h + (size_t)(row0 + half * 8) * FOUT + col;
#pragma unroll
            for (int r = 0; r < 8; ++r) hp[(size_t)r * FOUT] = acc[r];
        } else {
#pragma unroll
            for (int r = 0; r < 8; ++r) {
                const int m = row0 + r + half * 8;
                if (m < n) h[(size_t)m * FOUT + col] = acc[r];
            }
        }
    }
}

// ---------------------------------------------------------------------------
// Degree / normalization helpers
// ---------------------------------------------------------------------------
__global__ __launch_bounds__(256) void init_deg(float* __restrict__ deg, int n)
{
    int i = blockIdx.x * 256 + threadIdx.x;
    if (i < n) deg[i] = 1.0f;    // self-loop weight
}

__global__ __launch_bounds__(256) void deg_accum(
    const int* __restrict__ dst, const float* __restrict__ ew,
    float* __restrict__ deg, int e)
{
    int i = blockIdx.x * 256 + threadIdx.x;
    if (i < e) atomicAdd(&deg[dst[i]], ew[i]);
}

__global__ __launch_bounds__(256) void make_dinv(
    const float* __restrict__ deg, float* __restrict__ dinv,
    float* __restrict__ invdeg, int n)
{
    int i = blockIdx.x * 256 + threadIdx.x;
    if (i < n) {
        float d = deg[i];
        dinv[i]   = rsqrtf(d);
        invdeg[i] = 1.0f / d;
    }
}

__global__ __launch_bounds__(256) void zero_f32(float* __restrict__ p, long n)
{
    long i = (long)blockIdx.x * 256 + threadIdx.x;
    long stride = (long)gridDim.x * 256;
    for (; i < n; i += stride) p[i] = 0.0f;
}

// ---------------------------------------------------------------------------
// Edge scatter: agg[dst] += h[src] * (dinv[src]*ew*dinv[dst])
// One thread per edge, float4 row reads, global f32 atomics.
// ---------------------------------------------------------------------------
template <int F>
__global__ __launch_bounds__(256) void edge_scatter(
    const int* __restrict__ src, const int* __restrict__ dst,
    const float* __restrict__ ew, const float* __restrict__ dinv,
    const float* __restrict__ h, float* __restrict__ agg, int e)
{
    int i = blockIdx.x * 256 + threadIdx.x;
    if (i >= e) return;
    const int s = src[i];
    const int d = dst[i];
    const float* hs = h + (size_t)s * F;
    __builtin_prefetch(hs, 0, 0);                    // global_prefetch_b8
    const float norm = dinv[s] * ew[i] * dinv[d];
    float* ad = agg + (size_t)d * F;
#pragma unroll
    for (int j = 0; j < F; j += 4) {
        const float4 v = *reinterpret_cast<const float4*>(hs + j);
        atomicAdd(ad + j + 0, v.x * norm);
        atomicAdd(ad + j + 1, v.y * norm);
        atomicAdd(ad + j + 2, v.z * norm);
        atomicAdd(ad + j + 3, v.w * norm);
    }
}

// x_next = leaky(agg + h*invdeg + b), written in place over agg
template <int F>
__global__ __launch_bounds__(256) void finalize_layer(
    const float* __restrict__ h, const float* __restrict__ invdeg,
    const float* __restrict__ b, float* __restrict__ agg, int n)
{
    long idx = (long)blockIdx.x * 256 + threadIdx.x;
    long total = (long)n * F;
    if (idx >= total) return;
    int i = (int)(idx / F);          // constant divisor -> mul/shift
    int j = (int)(idx - (long)i * F);
    float v = agg[idx] + h[idx] * invdeg[i] + b[j];
    agg[idx] = leaky(v);
}

// ---------------------------------------------------------------------------
// Global sum pool over 48 features: LDS partials per block, then 48 atomics.
// ---------------------------------------------------------------------------
__global__ __launch_bounds__(256) void pool48(
    const float* __restrict__ x, float* __restrict__ g, int n)
{
    __shared__ float loc[48];
    if (threadIdx.x < 48) loc[threadIdx.x] = 0.0f;
    __syncthreads();
    int i = blockIdx.x * 256 + threadIdx.x;
    if (i < n) {
        const float* xr = x + (size_t)i * 48;
#pragma unroll
        for (int j = 0; j < 48; j += 4) {
            const float4 v = *reinterpret_cast<const float4*>(xr + j);
            atomicAdd(&loc[j + 0], v.x);   // ds_add_f32
            atomicAdd(&loc[j + 1], v.y);
            atomicAdd(&loc[j + 2], v.z);
            atomicAdd(&loc[j + 3], v.w);
        }
    }
    __syncthreads();
    if (threadIdx.x < 48) atomicAdd(&g[threadIdx.x], loc[threadIdx.x]);
}

// ---------------------------------------------------------------------------
// MLP head: [1,48] -> 32 -> 16 -> 2, single block of 64 threads
// ---------------------------------------------------------------------------
__global__ __launch_bounds__(64) void fc_head(
    const float* __restrict__ g,
    const float* __restrict__ w1, const float* __restrict__ b1,
    const float* __restrict__ w2, const float* __restrict__ b2,
    const float* __restrict__ w3, const float* __restrict__ b3,
    float* __restrict__ out)
{
    __shared__ float s0[48], s1[32], s2[16];
    const int t = threadIdx.x;
    if (t < 48) s0[t] = g[t];
    __syncthreads();
    if (t < 32) {
        float a = b1[t];
#pragma unroll
        for (int k = 0; k < 48; ++k) a += s0[k] * w1[k * 32 + t];
        s1[t] = leaky(a);
    }
    __syncthreads();
    if (t < 16) {
        float a = b2[t];
#pragma unroll
        for (int k = 0; k < 32; ++k) a += s1[k] * w2[k * 16 + t];
        s2[t] = leaky(a);
    }
    __syncthreads();
    if (t < 2) {
        float a = b3[t];
#pragma unroll
        for (int k = 0; k < 16; ++k) a += s2[k] * w3[k * 2 + t];
        out[t] = a;
    }
}

// ---------------------------------------------------------------------------
extern "C" void kernel_launch(void* const* d_in, const int* in_sizes, int n_in,
                              void* d_out, int out_size, void* d_ws, size_t ws_size,
                              hipStream_t stream)
{
    const float* xin  = (const float*)d_in[0];   // [N,4]
    const float* ew   = (const float*)d_in[1];   // [E]
    const int*   ei   = (const int*)  d_in[2];   // [2,E]
    const float* W1   = (const float*)d_in[3];
    const float* b1   = (const float*)d_in[4];
    const float* W2   = (const float*)d_in[5];
    const float* b2   = (const float*)d_in[6];
    const float* W3   = (const float*)d_in[7];
    const float* b3   = (const float*)d_in[8];
    const float* fcW1 = (const float*)d_in[9];
    const float* fcb1 = (const float*)d_in[10];
    const float* fcW2 = (const float*)d_in[11];
    const float* fcb2 = (const float*)d_in[12];
    const float* fcW3 = (const float*)d_in[13];
    const float* fcb3 = (const float*)d_in[14];

    const int N = in_sizes[0] / 4;
    const int E = in_sizes[1];
    const int* src = ei;
    const int* dst = ei + E;

    // Workspace layout (floats)
    float* ws = (float*)d_ws;
    const size_t NB = (size_t)N * 48;
    float* B0     = ws;                 // h buffer
    float* B1     = ws + NB;            // agg / x_cur ping-pong buffer
    float* deg    = ws + 2 * NB;
    float* dinv   = deg + N;
    float* invdeg = dinv + N;
    float* g48    = invdeg + N;

    const int nb256  = (N + 255) / 256;
    const int eb256  = (E + 255) / 256;
    const int mtiles = (N + 15) / 16;

    // Degrees (shared by all three layers)
    init_deg<<<nb256, 256, 0, stream>>>(deg, N);
    deg_accum<<<eb256, 256, 0, stream>>>(dst, ew, deg, E);
    make_dinv<<<nb256, 256, 0, stream>>>(deg, dinv, invdeg, N);

    // ---- Layer 1: fin=4, fout=12 ----
    {
        const int F = 12;
        gemm_wmma_f32<4, 12><<<dim3(mtiles, 1), 32, 0, stream>>>(xin, W1, B0, N);
        zero_f32<<<2048, 256, 0, stream>>>(B1, (long)N * F);
        edge_scatter<12><<<eb256, 256, 0, stream>>>(src, dst, ew, dinv, B0, B1, E);
        finalize_layer<12><<<(int)(((long)N * F + 255) / 256), 256, 0, stream>>>(
            B0, invdeg, b1, B1, N);
    }
    // ---- Layer 2: fin=12, fout=24 ----
    {
        const int F = 24;
        gemm_wmma_f32<12, 24><<<dim3(mtiles, 2), 32, 0, stream>>>(B1, W2, B0, N);
        zero_f32<<<2048, 256, 0, stream>>>(B1, (long)N * F);   // x_cur dead after GEMM
        edge_scatter<24><<<eb256, 256, 0, stream>>>(src, dst, ew, dinv, B0, B1, E);
        finalize_layer<24><<<(int)(((long)N * F + 255) / 256), 256, 0, stream>>>(
            B0, invdeg, b2, B1, N);
    }
    // ---- Layer 3: fin=24, fout=48 ----
    {
        const int F = 48;
        gemm_wmma_f32<24, 48><<<dim3(mtiles, 3), 32, 0, stream>>>(B1, W3, B0, N);
        zero_f32<<<2048, 256, 0, stream>>>(B1, (long)N * F);
        edge_scatter<48><<<eb256, 256, 0, stream>>>(src, dst, ew, dinv, B0, B1, E);
        finalize_layer<48><<<(int)(((long)N * F + 255) / 256), 256, 0, stream>>>(
            B0, invdeg, b3, B1, N);
    }

    // ---- Global sum pool + MLP head ----
    zero_f32<<<1, 256, 0, stream>>>(g48, 48);
    pool48<<<nb256, 256, 0, stream>>>(B1, g48, N);
    fc_head<<<1, 64, 0, stream>>>(g48, fcW1, fcb1, fcW2, fcb2, fcW3, fcb3,
                                  (float*)d_out);
}